// SequenceBucketPreprocessor_76596446757044
// MI455X (gfx1250) — compile-verified
//
#include <hip/hip_runtime.h>
#include <hip/hip_bf16.h>
#include <stdint.h>

// ---------------------------------------------------------------------------
// SequenceBucketPreprocessor for MI455X (gfx1250).
//
// out[b,t,s] = #{ k : features[b,t,slot_ids[s]] >= thresholds[s,k] }
// (thresholds sorted ascending, padded with +inf => this equals the
//  reference's first-hit index / bucket_nums fallback; bucket_nums unused).
//
// Memory-bound problem (~36.5 MB HBM traffic -> ~1.6us @ 23.3 TB/s). No
// matmul structure, so no WMMA; instead we use the CDNA5 Tensor Data Mover
// (tensor_load_to_lds + s_wait_tensorcnt) to stage all inputs into LDS and
// reuse each feature row 256x from LDS.
// ---------------------------------------------------------------------------

typedef __attribute__((ext_vector_type(4))) unsigned int v4u;
typedef __attribute__((ext_vector_type(4))) int          v4i;
typedef __attribute__((ext_vector_type(8))) int          v8i;

namespace {
constexpr int kB = 32, kT = 512, kF = 300;
constexpr int kS = 256, kK = 32;
constexpr int kRows        = kB * kT;            // 16384 (b,t) rows
constexpr int kRowsPerBlk  = 16;                 // rows staged per block
constexpr int kThreads     = kS;                 // 256 threads = 8 wave32
constexpr int kBlocks      = kRows / kRowsPerBlk; // 1024
}

// Build a 1-D TDM descriptor (D#) and issue TENSOR_LOAD_TO_LDS.
// Copies n_dwords consecutive DWORDs from gsrc to LDS byte address lds_addr.
// D# layout per CDNA5 ISA:
//   group0: [1:0]=count=1, [63:32]=lds_addr, [120:64]=global_addr, [127:126]=type(2)
//   group1: [17:16]=data_size(2 => 4B), [79:48]=tensor_dim0,
//           [127:112]=tile_dim0, tile_dim1/2 = 0 (1-D tile),
//           [207:160]=tensor_dim0_stride
//   groups 2/3 (and the extra zero group of the 6-arg builtin): zero.
__device__ __forceinline__ void tdm_load_dwords_to_lds(uint32_t lds_addr,
                                                       const void* gsrc,
                                                       uint32_t n_dwords) {
  const uint64_t ga = (uint64_t)(uintptr_t)gsrc;
  v4u g0;
  g0.x = 1u;                                                  // count=1, user D#
  g0.y = lds_addr;                                            // LDS dest (bytes)
  g0.z = (uint32_t)ga;                                        // global_addr[31:0]
  g0.w = (uint32_t)((ga >> 32) & 0x01FFFFFFu) | (2u << 30);   // addr[56:32] | type=2
  v8i g1;
  g1.s0 = (int)(2u << 16);                    // workgroup_mask=0, data_size=4B
  g1.s1 = (int)((n_dwords & 0xFFFFu) << 16);  // tensor_dim0[15:0]  @ bits 63:48
  g1.s2 = (int)(n_dwords >> 16);              // tensor_dim0[31:16]; tensor_dim1=0
  g1.s3 = (int)((n_dwords & 0xFFFFu) << 16);  // tile_dim0          @ bits 127:112
  g1.s4 = 0;                                  // tile_dim1=0, tile_dim2=0 (1-D)
  g1.s5 = (int)n_dwords;                      // tensor_dim0_stride[31:0]
  g1.s6 = 0;                                  // stride hi / dim1 stride lo
  g1.s7 = 0;
  v4i gz4 = {0, 0, 0, 0};
  v8i gz8 = {0, 0, 0, 0, 0, 0, 0, 0};
  // amdgpu-toolchain (clang-23) 6-arg form: (g0, g1, g2, g3, gX, cpol)
  __builtin_amdgcn_tensor_load_to_lds(g0, g1, gz4, gz4, gz8, 0);
}

__global__ void __launch_bounds__(kThreads)
SequenceBucketPreprocessor_kernel(const float* __restrict__ features,
                                  const float* __restrict__ thresholds,
                                  const int*   __restrict__ slot_ids,
                                  int*         __restrict__ out) {
  __shared__ float lds_thr[kS * kK];              // 32 KB   (whole table)
  __shared__ float lds_feat[kRowsPerBlk * kF];    // 18.75 KB (16 feature rows)
  __shared__ int   lds_slot[kS];                  // 1 KB

  const int row0 = blockIdx.x * kRowsPerBlk;

  // Wave 0 issues all three TDM DMAs; TDM ignores EXEC so a single-lane
  // branch issues each instruction exactly once per block.
  if (threadIdx.x == 0) {
    tdm_load_dwords_to_lds((uint32_t)(uintptr_t)lds_thr,  thresholds,
                           (uint32_t)(kS * kK));
    tdm_load_dwords_to_lds((uint32_t)(uintptr_t)lds_feat,
                           features + (size_t)row0 * kF,
                           (uint32_t)(kRowsPerBlk * kF));
    tdm_load_dwords_to_lds((uint32_t)(uintptr_t)lds_slot, slot_ids,
                           (uint32_t)kS);
    __builtin_amdgcn_s_wait_tensorcnt(0);         // TENSORcnt == 0
  }
  __syncthreads();                                // publish LDS to all 8 waves

  const int s = threadIdx.x;                      // bucket id, 0..255

  // Cache this bucket's 32 (inf-padded, sorted) thresholds in VGPRs.
  float th[kK];
#pragma unroll
  for (int k = 0; k < kK; ++k) th[k] = lds_thr[s * kK + k];
  const int slot = lds_slot[s];

#pragma unroll 4
  for (int r = 0; r < kRowsPerBlk; ++r) {
    const float v = lds_feat[r * kF + slot];
    int cnt = 0;
#pragma unroll
    for (int k = 0; k < kK; ++k) cnt += (v >= th[k]) ? 1 : 0;
    out[(size_t)(row0 + r) * kS + s] = cnt;       // coalesced 1 KB row store
  }
}

extern "C" void kernel_launch(void* const* d_in, const int* in_sizes, int n_in,
                              void* d_out, int out_size, void* d_ws, size_t ws_size,
                              hipStream_t stream) {
  (void)in_sizes; (void)n_in; (void)out_size; (void)d_ws; (void)ws_size;
  const float* features   = (const float*)d_in[0];
  const float* thresholds = (const float*)d_in[1];
  const int*   slot_ids   = (const int*)d_in[2];
  // d_in[3] (bucket_nums) is not needed: +inf threshold padding encodes it.
  int* out = (int*)d_out;

  hipLaunchKernelGGL(SequenceBucketPreprocessor_kernel,
                     dim3(kBlocks), dim3(kThreads), 0, stream,
                     features, thresholds, slot_ids, out);
}